// DigitCap_37589553774747
// MI455X (gfx1250) — compile-verified
//
#include <hip/hip_runtime.h>

// Problem constants (fixed by reference setup_inputs)
#define BB 32      // batch
#define NN 32768   // num primary caps
#define QQ 8       // dim primary
#define DD 10      // num digit caps
#define VV 16      // dim digit
#define SVEC (BB * DD * VV)   // 5120 floats for s / v / vsum

#define TILE 128              // n-tile staged in LDS
#define LROW (TILE * QQ + 4)  // padded row stride (floats); 1028 % 64 == 4 -> conflict-free

typedef float v2f __attribute__((ext_vector_type(2)));
typedef float v4f __attribute__((ext_vector_type(4)));
typedef float v8f __attribute__((ext_vector_type(8)));
typedef int   v4i __attribute__((vector_size(16)));   // matches async builtin's param type

typedef __attribute__((address_space(1))) v4i* gptr_v4i;
typedef __attribute__((address_space(3))) v4i* lptr_v4i;

__device__ __forceinline__ v8f wmma_f32_16x16x4(v2f a, v2f b, v8f c) {
    // D = A(16x4 f32) x B(4x16 f32) + C(16x16 f32)
    return __builtin_amdgcn_wmma_f32_16x16x4_f32(
        /*neg_a=*/false, a, /*neg_b=*/false, b,
        /*c_mod=*/(short)0, c, /*reuse_a=*/false, /*reuse_b=*/false);
}

// Pairwise dot of a WMMA accumulator (8 consecutive VGPRs) with preloaded v2f
// weights: even-aligned sub-vector pairs -> v_pk_fma_f32 without register movs.
__device__ __forceinline__ float dot8(v8f a, const v2f* w) {
    v2f p;
    p  = __builtin_shufflevector(a, a, 0, 1) * w[0];
    p += __builtin_shufflevector(a, a, 2, 3) * w[1];
    p += __builtin_shufflevector(a, a, 4, 5) * w[2];
    p += __builtin_shufflevector(a, a, 6, 7) * w[3];
    return p.x + p.y;
}

// One workgroup = 10 waves (one per digit d), processes n in [n0, n_end) in
// LDS-staged tiles of TILE. Per (d,n): u_hat[v=16,b=32] = W[d,n](16x8) @
// u[:,n,:]^T (8x32) via 4x V_WMMA_F32_16X16X4_F32.
// pass==0: uniform c = 1/D, accumulate straight in the WMMA C operand.
// pass>=1: logits = u_hat . Vsum, softmax over d via LDS, s += c * u_hat.
__global__ __launch_bounds__(DD * 32) void caps_pass(
    const float* __restrict__ u,        // [B, N, Q]
    const float* __restrict__ W,        // [D, N, V, Q]
    const float* __restrict__ vsum,     // [B, D, V]  (valid when pass >= 1)
    float* __restrict__ partials,       // [gridDim.x][B][D][V]
    int pass, int chunk)
{
    const int tid  = threadIdx.x;
    const int d    = tid >> 5;           // wave id == digit index, 0..9
    const int lane = tid & 31;
    const int half = lane >> 4;          // selects K pair / V half
    const int l16  = lane & 15;

    __shared__ float lds_u[BB * LROW];   // staged u tile: [b][nl*8+q], padded rows
    __shared__ float lds_e[2][DD][32];   // exp(logit), ping-pong parity

    // s accumulators in WMMA C/D layout:
    // lane L, vgpr j -> element (v = j + 8*(L/16), b = L%16 + 16*h)
    v8f acc0 = (v8f)0.0f;   // batch half h=0 (b = 0..15)
    v8f acc1 = (v8f)0.0f;   // batch half h=1 (b = 16..31)

    // Preload Vsum fragments matching the C layout (constant over n), as v2f
    // pairs so the logit dot maps onto v_pk_fma_f32 directly.
    v2f vs0[4], vs1[4];
    if (pass > 0) {
        const int vb = half * 8;
        const float* p0 = vsum + (l16 * DD + d) * VV + vb;
        const float* p1 = vsum + ((l16 + 16) * DD + d) * VV + vb;
#pragma unroll
        for (int j = 0; j < 4; ++j) {
            vs0[j] = *(const v2f*)(p0 + 2 * j);
            vs1[j] = *(const v2f*)(p1 + 2 * j);
        }
    }

    const int n0    = blockIdx.x * chunk;
    const int n_end = min(NN, n0 + chunk);
    int par = 0;

    for (int t0 = n0; t0 < n_end; t0 += TILE) {
        const int tl = min(TILE, n_end - t0);

        // ---- stage u tile into LDS (all 320 threads cooperate) ---------
        __syncthreads();   // previous tile fully consumed before overwrite
        {
            const int e4max = BB * tl * 2;            // float4 elements
            for (int i = tid; i < e4max; i += DD * 32) {
                const int b  = i / (tl * 2);
                const int r  = i - b * (tl * 2);
                const int nl = r >> 1;
                const int qq = (r & 1) * 4;
                const float* src = u + ((size_t)b * NN + (t0 + nl)) * QQ + qq;
                float* dst = &lds_u[b * LROW + nl * QQ + qq];
#if __has_builtin(__builtin_amdgcn_global_load_async_to_lds_b128)
                __builtin_amdgcn_global_load_async_to_lds_b128(
                    (gptr_v4i)(float*)src, (lptr_v4i)dst, 0, 0);
#else
                *(v4f*)dst = *(const v4f*)src;
#endif
            }
#if __has_builtin(__builtin_amdgcn_global_load_async_to_lds_b128)
#if __has_builtin(__builtin_amdgcn_s_wait_asynccnt)
            __builtin_amdgcn_s_wait_asynccnt(0);
#else
            asm volatile("s_wait_asynccnt 0x0" ::: "memory");
#endif
#endif
        }
        __syncthreads();   // tile visible to all waves

        for (int nl = 0; nl < tl; ++nl) {
            const int n = t0 + nl;

            // ---- A operand: W[d,n] contiguous 16x8 block (global, streamed)
            // A f32 16x4 layout: M = lane%16, K = 2*half + {0,1} (+4 per k-step)
            const float* wp = W + ((size_t)d * NN + n) * (VV * QQ);
            v2f a0 = *(const v2f*)(wp + l16 * QQ + half * 2);       // q in [0,4)
            v2f a1 = *(const v2f*)(wp + l16 * QQ + 4 + half * 2);   // q in [4,8)

            // ---- B operand: u[b,n,:]^T from LDS, 4x16 per (k-step, b-half)
            // B layout: N = lane%16, K = 2*half + {0,1}
            const float* ub0 = &lds_u[l16 * LROW + nl * QQ + half * 2];
            const float* ub1 = &lds_u[(l16 + 16) * LROW + nl * QQ + half * 2];
            v2f b00 = *(const v2f*)(ub0);       // h=0, q 0..3
            v2f b01 = *(const v2f*)(ub0 + 4);   // h=0, q 4..7
            v2f b10 = *(const v2f*)(ub1);       // h=1, q 0..3
            v2f b11 = *(const v2f*)(ub1 + 4);   // h=1, q 4..7

            if (pass == 0) {
                // c uniform = 1/D: accumulate u_hat sum in C, scale at end
                acc0 = wmma_f32_16x16x4(a0, b00, acc0);
                acc0 = wmma_f32_16x16x4(a1, b01, acc0);
                acc1 = wmma_f32_16x16x4(a0, b10, acc1);
                acc1 = wmma_f32_16x16x4(a1, b11, acc1);
            } else {
                v8f z = (v8f)0.0f;
                v8f uh0 = wmma_f32_16x16x4(a0, b00, z);
                uh0     = wmma_f32_16x16x4(a1, b01, uh0);
                v8f uh1 = wmma_f32_16x16x4(a0, b10, z);
                uh1     = wmma_f32_16x16x4(a1, b11, uh1);

                // logits: u_hat . Vsum over this lane's 8 v's (pk_fma pairs)
                float t0d = dot8(uh0, vs0);
                float t1d = dot8(uh1, vs1);
                // combine v halves across lane halves
                t0d += __shfl_xor(t0d, 16, 32);   // logit[b = l16]
                t1d += __shfl_xor(t1d, 16, 32);   // logit[b = l16 + 16]

                // each lane owns b == lane; write exp(logit) to LDS
                float lown = half ? t1d : t0d;
                float e = __expf(fminf(lown, 80.0f));
                lds_e[par][d][lane] = e;
                __syncthreads();

                // softmax denominator over d for own b; exchange across halves
                float den = 0.0f;
#pragma unroll
                for (int d2 = 0; d2 < DD; ++d2) den += lds_e[par][d2][lane];
                float den_o = __shfl_xor(den, 16, 32);
                float e_o   = __shfl_xor(e, 16, 32);
                float c_own = e * __builtin_amdgcn_rcpf(den);
                float c_oth = e_o * __builtin_amdgcn_rcpf(den_o);
                float c0 = half ? c_oth : c_own;   // c for b = l16
                float c1 = half ? c_own : c_oth;   // c for b = l16 + 16

                // s[b,d,v] += c * u_hat (whole-vector: pk_fma w/ scalar splat)
                acc0 += uh0 * c0;
                acc1 += uh1 * c1;
                par ^= 1;
            }
        }
    }

    // write deterministic per-workgroup partial: partials[wg][b][d][v]
    const float scale = (pass == 0) ? (1.0f / (float)DD) : 1.0f;
    v8f o0 = acc0 * scale;
    v8f o1 = acc1 * scale;
    float* pb = partials + (size_t)blockIdx.x * SVEC;
    const int vb = half * 8;
#pragma unroll
    for (int j = 0; j < 8; ++j)
        pb[(l16 * DD + d) * VV + vb + j] = o0[j];
#pragma unroll
    for (int j = 0; j < 8; ++j)
        pb[((l16 + 16) * DD + d) * VV + vb + j] = o1[j];
}

// Sum per-WG partials -> s[b,d,v]; squash over v (16 lanes); update vsum; write v.
__global__ __launch_bounds__(256) void caps_reduce(
    const float* __restrict__ partials, int G,
    float* __restrict__ vsum, float* __restrict__ out, int pass)
{
    const int idx = blockIdx.x * 256 + threadIdx.x;   // 0..SVEC-1; v = idx%16
    float s = 0.0f;
    for (int g = 0; g < G; ++g) s += partials[(size_t)g * SVEC + idx];

    // ||s||^2 over the 16-lane v-group (lanes 0-15 / 16-31 are separate caps)
    float nsq = s * s;
    nsq += __shfl_xor(nsq, 1, 32);
    nsq += __shfl_xor(nsq, 2, 32);
    nsq += __shfl_xor(nsq, 4, 32);
    nsq += __shfl_xor(nsq, 8, 32);

    float norm = sqrtf(nsq);
    float vout = s * nsq / ((1.0f + nsq) * (norm + 1e-7f));

    out[idx] = vout;
    if (pass == 0) vsum[idx] = vout;
    else           vsum[idx] += vout;
}

extern "C" void kernel_launch(void* const* d_in, const int* in_sizes, int n_in,
                              void* d_out, int out_size, void* d_ws, size_t ws_size,
                              hipStream_t stream) {
    const float* u = (const float*)d_in[0];   // primary_caps [B, N, Q] f32
    const float* W = (const float*)d_in[1];   // W [D, N, V, Q] f32
    float* out = (float*)d_out;               // v [B, D, V] f32

    float* ws = (float*)d_ws;
    float* vsum = ws;                         // [SVEC]
    float* partials = ws + SVEC;              // [G][SVEC]

    long cap = (long)(ws_size / sizeof(float)) - SVEC;
    int G = (int)(cap / SVEC);
    if (G > 256) G = 256;
    if (G < 1)   G = 1;
    const int chunk = (NN + G - 1) / G;

    for (int pass = 0; pass < 3; ++pass) {
        caps_pass<<<G, DD * 32, 0, stream>>>(u, W, vsum, partials, pass, chunk);
        caps_reduce<<<SVEC / 256, 256, 0, stream>>>(partials, G, vsum, out, pass);
    }
}